// TransformerBlockQuantum_65481071397340
// MI455X (gfx1250) — compile-verified
//
#include <hip/hip_runtime.h>
#include <hip/hip_bf16.h>

typedef __attribute__((ext_vector_type(2))) float v2f;
typedef __attribute__((ext_vector_type(8))) float v8f;

#define NB 8
#define NS 1024
#define NH 16
#define ND 8
#define NE 128
#define NF 512
#define SCALE 0.35355339059327373f    // 1/sqrt(8)
#define LOG2E 1.4426950408889634f     // log2(e)

__device__ __forceinline__ v8f wmma4(v2f a, v2f b, v8f c) {
  // D = A(16x4,f32) * B(4x16,f32) + C(16x16,f32)
  return __builtin_amdgcn_wmma_f32_16x16x4_f32(false, a, false, b, (short)0, c,
                                               false, false);
}

// ---------------- Kernel 1: quantum ring encoding -> q[B,H,S,D] --------------
__global__ void k_qz(const float* __restrict__ x, const float* __restrict__ rx,
                     float* __restrict__ q) {
  int idx = blockIdx.x * blockDim.x + threadIdx.x;  // over B*S*H = 131072
  const float* xp = x + (size_t)idx * ND;           // x[b,s,h,:]
  float c[ND];
#pragma unroll
  for (int d = 0; d < ND; ++d) c[d] = __cosf(xp[d] + rx[d]);
  float o[ND];
  float P = c[0];
#pragma unroll
  for (int d = 1; d < ND; ++d) { P *= c[d]; o[d] = P; }
  float z0 = c[1];
#pragma unroll
  for (int d = 2; d < ND; ++d) z0 *= c[d];
  o[0] = z0;
  int h = idx & 15;
  int s = (idx >> 4) & (NS - 1);
  int b = idx >> 14;
  float* qp = q + ((size_t)(b * NH + h) * NS + s) * ND;
#pragma unroll
  for (int d = 0; d < ND; ++d) qp[d] = o[d];
}

// ---------------- Kernel 2: attention, wave per (b,h,qtile) ------------------
// q==k==v, |score| <= 8/sqrt(8) so softmax needs no max subtraction.
// log2(e) is folded into the Q pre-scale so the QK^T WMMA emits scores in the
// log2 domain and p = exp2(s) is a single v_exp_f32 per register.
// Denominator accumulated by the matrix unit via an all-ones B column (col 8).
// The K tile (already in registers) is staged to LDS as V^T with extension
// columns, so the PV B operand is a branch-free ds_load_2addr.
__global__ void k_attn(const float* __restrict__ q, float* __restrict__ ctx) {
  __shared__ float Pbuf[8][16][18];  // per-wave P tile, stride padded for b64
  __shared__ float Vbuf[8][16][18];  // per-wave [key][dim0..7 | 1 | 0...]
  int wave = threadIdx.x >> 5;
  int lane = threadIdx.x & 31;
  int half = lane >> 4;
  int l16 = lane & 15;
  int wid = blockIdx.x * 8 + wave;  // 8192 waves total
  int qt = wid & 63;                // 64 query tiles of 16
  int bh = wid >> 6;                // 128 (b,h) pairs
  const float* qbase = q + (size_t)bh * NS * ND;

  float(*P)[18] = Pbuf[wave];
  float(*Vt)[18] = Vbuf[wave];

  // Extension columns of the extended-V tile, written once:
  // col 8 = 1 (softmax denominator), cols 9..15 = 0.
  Vt[l16][8] = 1.f;
#pragma unroll
  for (int c = 9; c < 16; ++c) Vt[l16][c] = 0.f;

  // A operand (Q tile, 16x8), pre-scaled by (1/sqrt(8))*log2(e).
  // A layout: row=l16, k=2*half+{0,1} (+4)
  const float* qrow = qbase + (size_t)(qt * 16 + l16) * ND;
  const float ascale = SCALE * LOG2E;
  v2f a0, a1;
  a0.x = qrow[2 * half + 0] * ascale;
  a0.y = qrow[2 * half + 1] * ascale;
  a1.x = qrow[4 + 2 * half + 0] * ascale;
  a1.y = qrow[4 + 2 * half + 1] * ascale;

  v8f acc = {};  // C layout: cols 0..7 = ctx, col 8 = softmax denominator

  for (int kt = 0; kt < 64; ++kt) {
    // B operand (K tile^T): B(k,n)=K[n][k], n=l16, k=2*half+{0,1} (+4)
    const float* krow = qbase + (size_t)(kt * 16 + l16) * ND;
    v2f b0, b1;
    b0.x = krow[2 * half + 0];
    b0.y = krow[2 * half + 1];
    b1.x = krow[4 + 2 * half + 0];
    b1.y = krow[4 + 2 * half + 1];

    // Stage the K tile into LDS as V^T[key][dim] (aligned b64 stores).
    *(v2f*)&Vt[l16][2 * half] = b0;
    *(v2f*)&Vt[l16][4 + 2 * half] = b1;

    v8f s = {};
    s = wmma4(a0, b0, s);
    s = wmma4(a1, b1, s);

    // p = exp2(score*log2e): one v_exp_f32 per register, no pre-multiply.
#pragma unroll
    for (int v = 0; v < 8; ++v) {
      P[v + 8 * half][l16] = __builtin_amdgcn_exp2f(s[v]);
    }
    // Same-wave LDS store->load: DS pipeline is in-order; the compiler
    // inserts precise s_wait_dscnt for the load-data consumers.

    // ctx/denom += P(16x16) @ [V | 1 | 0](16x16), K=16 -> 4 chained WMMAs.
#pragma unroll
    for (int kc = 0; kc < 4; ++kc) {
      int kr = 4 * kc + 2 * half;
      v2f pa = *(const v2f*)&P[l16][kr];  // A: row=l16, k=kr,kr+1
      v2f pb;                             // B: rows kr,kr+1, col n=l16
      pb.x = Vt[kr][l16];
      pb.y = Vt[kr + 1][l16];
      acc = wmma4(pa, pb, acc);
    }
  }

  // Broadcast denominator (col 8) within each 16-lane half, normalize, write
  // ctx in [B*S, E] layout (col = h*8 + d).
  float rden[8];
#pragma unroll
  for (int v = 0; v < 8; ++v)
    rden[v] = __builtin_amdgcn_rcpf(__shfl(acc[v], 16 * half + 8, 32));

  if (l16 < 8) {
    int b = bh >> 4, h = bh & 15;
#pragma unroll
    for (int v = 0; v < 8; ++v) {
      int srow = qt * 16 + v + 8 * half;
      ctx[((size_t)(b * NS + srow)) * NE + h * ND + l16] = acc[v] * rden[v];
    }
  }
}

// ------- Kernel 3: attn_out = ctx @ w_comb^T + b; x1 = LN(x + attn_out) ------
// Block = 256 threads (8 waves) handles 16 rows x 128 cols; wave w -> cols 16w.
__global__ void k_comb_ln(const float* __restrict__ ctx,
                          const float* __restrict__ w_comb,
                          const float* __restrict__ b_comb,
                          const float* __restrict__ x,
                          const float* __restrict__ g1,
                          const float* __restrict__ be1,
                          float* __restrict__ x1) {
  __shared__ float As[16][132];
  __shared__ float Os[16][132];
  int t = threadIdx.x;
  int r0 = blockIdx.x * 16;
  for (int i = t; i < 16 * NE; i += 256) {
    int r = i >> 7, c = i & 127;
    As[r][c] = ctx[(size_t)(r0 + r) * NE + c];
  }
  __syncthreads();

  int wave = t >> 5, lane = t & 31, half = lane >> 4, l16 = lane & 15;
  int cb = wave * 16;
  v8f acc = {};
  for (int kc = 0; kc < 32; ++kc) {  // K = 128
    int k = 4 * kc + 2 * half;
    v2f a;
    a.x = As[l16][k];
    a.y = As[l16][k + 1];
    const float* wr = w_comb + (size_t)(cb + l16) * NE + k;
    v2f b;
    b.x = wr[0];
    b.y = wr[1];
    acc = wmma4(a, b, acc);
  }
#pragma unroll
  for (int v = 0; v < 8; ++v) {
    int r = v + 8 * half;
    Os[r][cb + l16] =
        acc[v] + b_comb[cb + l16] + x[(size_t)(r0 + r) * NE + cb + l16];
  }
  __syncthreads();

  // LayerNorm: 16 threads per row (same 16-lane half), 8 cols each.
  int row = t >> 4, j = t & 15;
  float s1 = 0.f, s2 = 0.f;
#pragma unroll
  for (int c = 0; c < 8; ++c) {
    float v = Os[row][j * 8 + c];
    s1 += v;
    s2 += v * v;
  }
#pragma unroll
  for (int off = 1; off < 16; off <<= 1) {
    s1 += __shfl_xor(s1, off, 32);
    s2 += __shfl_xor(s2, off, 32);
  }
  float mu = s1 * (1.f / 128.f);
  float var = s2 * (1.f / 128.f) - mu * mu;
  float rs = __builtin_amdgcn_rsqf(var + 1e-5f);
#pragma unroll
  for (int c = 0; c < 8; ++c) {
    int cc = j * 8 + c;
    x1[(size_t)(r0 + row) * NE + cc] = (Os[row][cc] - mu) * rs * g1[cc] + be1[cc];
  }
}

// ------- Kernel 4: quantum FFN + residual + LN2 -> out -----------------------
__global__ void k_ffn(const float* __restrict__ x1, const float* __restrict__ ry,
                      const float* __restrict__ w1, const float* __restrict__ b1,
                      const float* __restrict__ w2, const float* __restrict__ b2,
                      const float* __restrict__ g2, const float* __restrict__ be2,
                      float* __restrict__ out) {
  __shared__ float zfs[16][8];
  __shared__ float Hs[16][516];
  __shared__ float Os[16][132];
  int t = threadIdx.x;
  int r0 = blockIdx.x * 16;

  if (t < 128) {  // zf = cos(x1[:, :8]) * cos(ry)
    int r = t >> 3, i = t & 7;
    zfs[r][i] = __cosf(x1[(size_t)(r0 + r) * NE + i]) * __cosf(ry[i]);
  }
  __syncthreads();

  // h = relu(zf @ w1^T + b1): K=8, plain VALU into LDS.
  for (int i = t; i < 16 * NF; i += 256) {
    int r = i >> 9, f = i & 511;
    float a = b1[f];
    const float* wr = w1 + (size_t)f * 8;
#pragma unroll
    for (int k = 0; k < 8; ++k) a += zfs[r][k] * wr[k];
    Hs[r][f] = fmaxf(a, 0.f);
  }
  __syncthreads();

  // ffn_out = h @ w2^T : K=512 -> 128 chained WMMAs per wave.
  int wave = t >> 5, lane = t & 31, half = lane >> 4, l16 = lane & 15;
  int cb = wave * 16;
  v8f acc = {};
  for (int kc = 0; kc < 128; ++kc) {
    int k = 4 * kc + 2 * half;
    v2f a;
    a.x = Hs[l16][k];
    a.y = Hs[l16][k + 1];
    const float* wr = w2 + (size_t)(cb + l16) * NF + k;
    v2f b;
    b.x = wr[0];
    b.y = wr[1];
    acc = wmma4(a, b, acc);
  }
#pragma unroll
  for (int v = 0; v < 8; ++v) {
    int r = v + 8 * half;
    Os[r][cb + l16] =
        acc[v] + b2[cb + l16] + x1[(size_t)(r0 + r) * NE + cb + l16];
  }
  __syncthreads();

  int row = t >> 4, j = t & 15;
  float s1 = 0.f, s2 = 0.f;
#pragma unroll
  for (int c = 0; c < 8; ++c) {
    float v = Os[row][j * 8 + c];
    s1 += v;
    s2 += v * v;
  }
#pragma unroll
  for (int off = 1; off < 16; off <<= 1) {
    s1 += __shfl_xor(s1, off, 32);
    s2 += __shfl_xor(s2, off, 32);
  }
  float mu = s1 * (1.f / 128.f);
  float var = s2 * (1.f / 128.f) - mu * mu;
  float rs = __builtin_amdgcn_rsqf(var + 1e-5f);
#pragma unroll
  for (int c = 0; c < 8; ++c) {
    int cc = j * 8 + c;
    out[(size_t)(r0 + row) * NE + cc] = (Os[row][cc] - mu) * rs * g2[cc] + be2[cc];
  }
}

extern "C" void kernel_launch(void* const* d_in, const int* in_sizes, int n_in,
                              void* d_out, int out_size, void* d_ws,
                              size_t ws_size, hipStream_t stream) {
  const float* x      = (const float*)d_in[0];
  const float* rx     = (const float*)d_in[1];
  const float* ry     = (const float*)d_in[2];
  const float* w_comb = (const float*)d_in[3];
  const float* b_comb = (const float*)d_in[4];
  const float* w1     = (const float*)d_in[5];
  const float* b1     = (const float*)d_in[6];
  const float* w2     = (const float*)d_in[7];
  const float* b2     = (const float*)d_in[8];
  const float* g1     = (const float*)d_in[9];
  const float* be1    = (const float*)d_in[10];
  const float* g2     = (const float*)d_in[11];
  const float* be2    = (const float*)d_in[12];
  float* out = (float*)d_out;

  float* ws  = (float*)d_ws;
  float* q   = ws;                          // B*H*S*D  = 1,048,576 floats
  float* ctx = ws + (size_t)1048576;        // B*S*E    = 1,048,576 floats
  float* x1  = ws + (size_t)2097152;        // B*S*E    = 1,048,576 floats

  k_qz<<<512, 256, 0, stream>>>(x, rx, q);                       // 131072 thr
  k_attn<<<1024, 256, 0, stream>>>(q, ctx);                      // 8192 waves
  k_comb_ln<<<512, 256, 0, stream>>>(ctx, w_comb, b_comb, x, g1, be1, x1);
  k_ffn<<<512, 256, 0, stream>>>(x1, ry, w1, b1, w2, b2, g2, be2, out);
}